// topk_crossEntrophy_88270167867970
// MI455X (gfx1250) — compile-verified
//
#include <hip/hip_runtime.h>
#include <stdint.h>

typedef __attribute__((ext_vector_type(2))) float v2f;
typedef __attribute__((ext_vector_type(8))) float v8f;

#define N_ROWS 262144
#define N_COLS 128
#define TOPK   183500          // int(0.7 * 262144)
#define TILES  (N_ROWS / 16)   // 16384

// ---------------------------------------------------------------------------
// Kernel 1: per-row NLL loss via logsumexp; row-sum reduction done with
// V_WMMA_F32_16X16X4_F32 against a ones B-matrix (16 rows per wave).
// Four independent accumulators break the D->C dependency chain so the
// matrix pipe can issue at throughput while v_exp co-executes.
// Output: order-preserving uint32 keys (larger key == larger loss).
// ---------------------------------------------------------------------------
__global__ __launch_bounds__(256) void loss_kernel(const float* __restrict__ inp,
                                                   const int* __restrict__ tgt,
                                                   uint32_t* __restrict__ keys) {
    const unsigned lane = threadIdx.x & 31u;
    const unsigned m    = lane & 15u;    // row within tile
    const unsigned h    = lane >> 4;     // K-pair half (A-matrix layout)
    const unsigned w    = blockIdx.x * (blockDim.x >> 5) + (threadIdx.x >> 5);
    const unsigned numWaves = gridDim.x * (blockDim.x >> 5);

    const v2f ones = {1.0f, 1.0f};

    for (unsigned tile = w; tile < TILES; tile += numWaves) {
        const unsigned row = tile * 16u + m;
        const float* rp = inp + (size_t)row * N_COLS + 2u * h;

        // prefetch next tile's stream
        if (tile + numWaves < TILES) {
            __builtin_prefetch(inp + (size_t)(tile + numWaves) * 16u * N_COLS
                                   + (size_t)m * N_COLS + 2u * h, 0, 0);
        }

        // Phase 1: load row half (64 floats/lane, interleaved K pairs), row max
        v2f v[32];
        float mx = -3.402823466e38f;
        #pragma unroll
        for (int i = 0; i < 32; ++i) {
            v[i] = *(const v2f*)(rp + 4 * i);
            mx = fmaxf(mx, fmaxf(v[i].x, v[i].y));
        }
        mx = fmaxf(mx, __shfl_xor(mx, 16, 32));   // full row max on both halves

        // Phase 2: sum(exp(x - max)) for 16 rows at once via WMMA with B = ones.
        // 4 independent accumulator chains (8 WMMAs each).
        v8f c0 = {0.f, 0.f, 0.f, 0.f, 0.f, 0.f, 0.f, 0.f};
        v8f c1 = {0.f, 0.f, 0.f, 0.f, 0.f, 0.f, 0.f, 0.f};
        v8f c2 = {0.f, 0.f, 0.f, 0.f, 0.f, 0.f, 0.f, 0.f};
        v8f c3 = {0.f, 0.f, 0.f, 0.f, 0.f, 0.f, 0.f, 0.f};
        #pragma unroll
        for (int i = 0; i < 8; ++i) {
            v2f a0, a1, a2, a3;
            a0.x = __expf(v[4*i + 0].x - mx);  a0.y = __expf(v[4*i + 0].y - mx);
            a1.x = __expf(v[4*i + 1].x - mx);  a1.y = __expf(v[4*i + 1].y - mx);
            a2.x = __expf(v[4*i + 2].x - mx);  a2.y = __expf(v[4*i + 2].y - mx);
            a3.x = __expf(v[4*i + 3].x - mx);  a3.y = __expf(v[4*i + 3].y - mx);
            c0 = __builtin_amdgcn_wmma_f32_16x16x4_f32(false, a0, false, ones,
                                                       (short)0, c0, false, false);
            c1 = __builtin_amdgcn_wmma_f32_16x16x4_f32(false, a1, false, ones,
                                                       (short)0, c1, false, false);
            c2 = __builtin_amdgcn_wmma_f32_16x16x4_f32(false, a2, false, ones,
                                                       (short)0, c2, false, false);
            c3 = __builtin_amdgcn_wmma_f32_16x16x4_f32(false, a3, false, ones,
                                                       (short)0, c3, false, false);
        }
        v8f c = (c0 + c1) + (c2 + c3);

        // c[r] on lanes 0-15 holds S_r, on lanes 16-31 holds S_{8+r}
        // (replicated across columns). Gather own row's sum to lanes h==0.
        const int src = (m < 8u) ? 0 : 16;
        float S = 0.f;
        #pragma unroll
        for (int r = 0; r < 8; ++r) {
            float t = __shfl(c[r], src, 32);
            S = ((int)(m & 7u) == r) ? t : S;
        }

        if (h == 0u) {
            int   t    = tgt[row];
            float xt   = inp[(size_t)row * N_COLS + t];   // cache hit (row just streamed)
            float loss = mx + __logf(S) - xt;
            uint32_t b   = __float_as_uint(loss);
            uint32_t key = (b & 0x80000000u) ? ~b : (b | 0x80000000u);
            keys[row] = key;
        }
    }
}

// ---------------------------------------------------------------------------
// Radix-select (MSB-first) for the k-th largest key.
// ---------------------------------------------------------------------------
__global__ void init_kernel(uint32_t* hist, uint32_t* state) {
    if (threadIdx.x < 256u) hist[threadIdx.x] = 0u;
    if (threadIdx.x == 0u) { state[0] = 0u; state[1] = (uint32_t)TOPK; }
}

__global__ __launch_bounds__(256) void hist_kernel(const uint32_t* __restrict__ keys,
                                                   uint32_t* __restrict__ hist,
                                                   const uint32_t* __restrict__ state,
                                                   int pass) {
    __shared__ uint32_t lh[256];
    lh[threadIdx.x] = 0u;
    __syncthreads();

    const uint32_t prefix   = state[0];
    const uint32_t maskHigh = (pass == 3) ? 0u : (0xFFFFFFFFu << (8 * (pass + 1)));
    const int shift = 8 * pass;

    for (unsigned i = blockIdx.x * blockDim.x + threadIdx.x; i < N_ROWS;
         i += gridDim.x * blockDim.x) {
        uint32_t key = keys[i];
        if ((key & maskHigh) == prefix)
            atomicAdd(&lh[(key >> shift) & 0xFFu], 1u);
    }
    __syncthreads();
    if (lh[threadIdx.x]) atomicAdd(&hist[threadIdx.x], lh[threadIdx.x]);
}

__global__ void scan_kernel(uint32_t* hist, uint32_t* state, int pass) {
    // single-thread: pick digit d with count(>d) < kRem <= count(>=d)
    uint32_t kRem = state[1];
    uint32_t cumAbove = 0u;
    int sel = 0;
    for (int d = 255; d >= 0; --d) {
        uint32_t hv = hist[d];
        if (cumAbove + hv >= kRem) { sel = d; break; }
        cumAbove += hv;
    }
    state[0] |= ((uint32_t)sel) << (8 * pass);
    state[1]  = kRem - cumAbove;
    for (int d = 0; d < 256; ++d) hist[d] = 0u;   // ready for next pass
}

// ---------------------------------------------------------------------------
// Deterministic sum of values strictly above the threshold key.
// ---------------------------------------------------------------------------
__global__ __launch_bounds__(256) void sum_kernel(const uint32_t* __restrict__ keys,
                                                  const uint32_t* __restrict__ state,
                                                  float* __restrict__ partials) {
    __shared__ float sm[256];
    const uint32_t T = state[0];
    float acc = 0.f;
    for (unsigned i = blockIdx.x * blockDim.x + threadIdx.x; i < N_ROWS;
         i += gridDim.x * blockDim.x) {
        uint32_t key = keys[i];
        if (key > T) {
            uint32_t b = (key & 0x80000000u) ? (key ^ 0x80000000u) : ~key;
            acc += __uint_as_float(b);
        }
    }
    sm[threadIdx.x] = acc;
    __syncthreads();
    for (int s = 128; s > 0; s >>= 1) {
        if ((int)threadIdx.x < s) sm[threadIdx.x] += sm[threadIdx.x + s];
        __syncthreads();
    }
    if (threadIdx.x == 0u) partials[blockIdx.x] = sm[0];
}

__global__ __launch_bounds__(256) void final_kernel(const float* __restrict__ partials,
                                                    const uint32_t* __restrict__ state,
                                                    float* __restrict__ out) {
    __shared__ float sm[256];
    sm[threadIdx.x] = partials[threadIdx.x] + partials[threadIdx.x + 256];
    __syncthreads();
    for (int s = 128; s > 0; s >>= 1) {
        if ((int)threadIdx.x < s) sm[threadIdx.x] += sm[threadIdx.x + s];
        __syncthreads();
    }
    if (threadIdx.x == 0u) {
        uint32_t T    = state[0];
        uint32_t kRem = state[1];                 // ties at the threshold to include
        uint32_t b = (T & 0x80000000u) ? (T ^ 0x80000000u) : ~T;
        float vT = __uint_as_float(b);
        out[0] = (sm[0] + (float)kRem * vT) / (float)TOPK;
    }
}

// ---------------------------------------------------------------------------
extern "C" void kernel_launch(void* const* d_in, const int* in_sizes, int n_in,
                              void* d_out, int out_size, void* d_ws, size_t ws_size,
                              hipStream_t stream) {
    const float* inp = (const float*)d_in[0];
    const int*   tgt = (const int*)d_in[1];
    float*       out = (float*)d_out;

    uint8_t* ws = (uint8_t*)d_ws;
    uint32_t* keys     = (uint32_t*)ws;                                   // N*4 bytes
    uint32_t* hist     = (uint32_t*)(ws + (size_t)N_ROWS * 4);            // 1 KB
    uint32_t* state    = (uint32_t*)(ws + (size_t)N_ROWS * 4 + 1024);     // 8 B
    float*    partials = (float*)   (ws + (size_t)N_ROWS * 4 + 1024 + 64);// 2 KB

    loss_kernel<<<1024, 256, 0, stream>>>(inp, tgt, keys);
    init_kernel<<<1, 256, 0, stream>>>(hist, state);
    for (int p = 3; p >= 0; --p) {
        hist_kernel<<<512, 256, 0, stream>>>(keys, hist, state, p);
        scan_kernel<<<1, 1, 0, stream>>>(hist, state, p);
    }
    sum_kernel<<<512, 256, 0, stream>>>(keys, state, partials);
    final_kernel<<<1, 256, 0, stream>>>(partials, state, out);
}